// GraphConv_65970697666703
// MI455X (gfx1250) — compile-verified
//
#include <hip/hip_runtime.h>

#define IN_F   256
#define OUT_F  128
#define ROWS_PB 64          // rows of feat per block (4 strips of 16)
#define FP     260          // LDS pitch for feat tile (pad 256 -> 260, 4-bank skew)
#define WP     132          // LDS pitch for W^T (pad 128 -> 132, 4-bank skew)

typedef __attribute__((ext_vector_type(2))) float v2f;
typedef __attribute__((ext_vector_type(8))) float v8f;

// -------------------------------------------------------------------------
// h[N,128] = feat[N,256] @ W[128,256]^T + b     (exact f32 via WMMA 16x16x4)
// Block: 256 threads = 8 wave32s. Wave w owns output cols [16w, 16w+16).
// Each wave accumulates 4 row-strips of 16 -> 64 rows/block.
// -------------------------------------------------------------------------
__launch_bounds__(256, 1)
__global__ void gemm_bias_wmma(const float* __restrict__ feat,
                               const float* __restrict__ W,
                               const float* __restrict__ bias,
                               float* __restrict__ h,
                               int N) {
    __shared__ float sW[IN_F * WP];      // W transposed: sW[k*WP + col]
    __shared__ float sF[ROWS_PB * FP];   // feat tile:    sF[r*FP + k]

    const int t       = threadIdx.x;
    const int rowBase = blockIdx.x * ROWS_PB;
    const bool full   = (rowBase + ROWS_PB) <= N;   // uniform: interior block?

    // ---- stage W^T into LDS (coalesced float4 global reads) ----
#pragma unroll
    for (int i = 0; i < 32; ++i) {
        int id  = t + i * 256;           // float4 id in [0, 8192)
        int col = id >> 6;               // 64 float4 per W row
        int k4  = (id & 63) << 2;
        float4 w4 = *(const float4*)(W + col * IN_F + k4);
        sW[(k4 + 0) * WP + col] = w4.x;
        sW[(k4 + 1) * WP + col] = w4.y;
        sW[(k4 + 2) * WP + col] = w4.z;
        sW[(k4 + 3) * WP + col] = w4.w;
    }
    // ---- stage 64x256 feat tile into LDS ----
    if (full) {
#pragma unroll
        for (int i = 0; i < 16; ++i) {
            int id = t + i * 256;        // float4 id in [0, 4096)
            int r  = id >> 6;
            int k4 = (id & 63) << 2;
            float4 f4 = *(const float4*)(feat + (size_t)(rowBase + r) * IN_F + k4);
            *(float4*)(sF + r * FP + k4) = f4;
        }
    } else {
#pragma unroll
        for (int i = 0; i < 16; ++i) {
            int id = t + i * 256;
            int r  = id >> 6;
            int k4 = (id & 63) << 2;
            int grow = rowBase + r;
            float4 f4 = {0.f, 0.f, 0.f, 0.f};
            if (grow < N) f4 = *(const float4*)(feat + (size_t)grow * IN_F + k4);
            *(float4*)(sF + r * FP + k4) = f4;
        }
    }
    __syncthreads();

    const int lane    = t & 31;
    const int waveId  = t >> 5;
    const int colBase = waveId << 4;          // this wave's 16 output cols
    const int n       = lane & 15;            // M for A, N for B/C
    const int kb      = (lane >> 4) << 1;     // lanes 0-15: K+{0,1}; 16-31: K+{2,3}

    v8f acc[4];
#pragma unroll
    for (int s = 0; s < 4; ++s) acc[s] = (v8f){0.f,0.f,0.f,0.f,0.f,0.f,0.f,0.f};

    for (int k = 0; k < IN_F; k += 4) {
        v2f B;                                 // B[k..k+3, colBase..+16)
        B.x = sW[(k + kb)     * WP + colBase + n];
        B.y = sW[(k + kb + 1) * WP + colBase + n];
#pragma unroll
        for (int s = 0; s < 4; ++s) {          // 4 row strips reuse B
            const float* fr = sF + (s * 16 + n) * FP + k + kb;
            v2f A;
            A.x = fr[0];
            A.y = fr[1];
            acc[s] = __builtin_amdgcn_wmma_f32_16x16x4_f32(
                false, A, false, B, (short)0, acc[s], false, false);
        }
    }

    // ---- bias add + store (C layout: VGPR j -> M = j + 8*(lane>=16), N = lane&15)
    const float bv = bias[colBase + n];
    if (full) {
#pragma unroll
        for (int s = 0; s < 4; ++s) {
            int mBase = rowBase + s * 16 + ((lane >> 4) << 3);
#pragma unroll
            for (int j = 0; j < 8; ++j)
                h[(size_t)(mBase + j) * OUT_F + colBase + n] = acc[s][j] + bv;
        }
    } else {
#pragma unroll
        for (int s = 0; s < 4; ++s) {
            int mBase = rowBase + s * 16 + ((lane >> 4) << 3);
#pragma unroll
            for (int j = 0; j < 8; ++j) {
                int grow = mBase + j;
                if (grow < N)
                    h[(size_t)grow * OUT_F + colBase + n] = acc[s][j] + bv;
            }
        }
    }
}

// -------------------------------------------------------------------------
// out[r] += val * h[c]  per edge.  One wave32 per edge: 32 lanes x float4
// = 128 feats, L2-resident gather + hardware fp32 atomics scatter.
// Edge descriptors are wave-uniform -> force SGPR so they go down the
// scalar/constant-cache path (KMcnt), keeping VMEM for data.
// -------------------------------------------------------------------------
__launch_bounds__(256)
__global__ void spmm_atomic(const int* __restrict__ erow,
                            const int* __restrict__ ecol,
                            const float* __restrict__ eval,
                            const float* __restrict__ h,
                            float* __restrict__ out,
                            int E) {
    const int lane    = threadIdx.x & 31;
    const int wInBlk  = __builtin_amdgcn_readfirstlane(threadIdx.x >> 5);
    const int wavesPB = blockDim.x >> 5;
    int       e       = blockIdx.x * wavesPB + wInBlk;        // scalar
    const int nwaves  = gridDim.x * wavesPB;                  // scalar
    for (; e < E; e += nwaves) {
        const int   r = erow[e];             // s_load (uniform)
        const int   c = ecol[e];
        const float v = eval[e];
        const float4 hv = *(const float4*)(h + (size_t)c * OUT_F + (lane << 2));
        float* o = out + (size_t)r * OUT_F + (lane << 2);
        unsafeAtomicAdd(o + 0, hv.x * v);
        unsafeAtomicAdd(o + 1, hv.y * v);
        unsafeAtomicAdd(o + 2, hv.z * v);
        unsafeAtomicAdd(o + 3, hv.w * v);
    }
}

__launch_bounds__(256)
__global__ void relu_kernel(float* __restrict__ out, long n4) {
    long i      = (long)blockIdx.x * blockDim.x + threadIdx.x;
    long stride = (long)gridDim.x * blockDim.x;
    float4* p = (float4*)out;
    for (; i < n4; i += stride) {
        float4 v = p[i];
        v.x = fmaxf(v.x, 0.f);
        v.y = fmaxf(v.y, 0.f);
        v.z = fmaxf(v.z, 0.f);
        v.w = fmaxf(v.w, 0.f);
        p[i] = v;
    }
}

extern "C" void kernel_launch(void* const* d_in, const int* in_sizes, int n_in,
                              void* d_out, int out_size, void* d_ws, size_t ws_size,
                              hipStream_t stream) {
    const float* feat = (const float*)d_in[0];
    const int*   erow = (const int*)  d_in[1];
    const int*   ecol = (const int*)  d_in[2];
    const float* eval = (const float*)d_in[3];
    const float* W    = (const float*)d_in[4];
    const float* bias = (const float*)d_in[5];

    const int N = in_sizes[0] / IN_F;     // 100000
    const int E = in_sizes[1];            // 3200000

    float* h   = (float*)d_ws;            // [N, 128] projected features
    float* out = (float*)d_out;           // [N, 128]

    // 1) project + bias (WMMA f32)
    int gemmBlocks = (N + ROWS_PB - 1) / ROWS_PB;
    gemm_bias_wmma<<<gemmBlocks, 256, 0, stream>>>(feat, W, bias, h, N);

    // 2) zero the accumulator (graph-capture-safe)
    hipMemsetAsync(out, 0, (size_t)N * OUT_F * sizeof(float), stream);

    // 3) edge scatter-add
    spmm_atomic<<<4096, 256, 0, stream>>>(erow, ecol, eval, h, out, E);

    // 4) ReLU
    long n4 = ((long)N * OUT_F) >> 2;
    relu_kernel<<<1024, 256, 0, stream>>>(out, n4);
}